// SIA_POSE_SIMPLE_DEC_ROI_39848706573700
// MI455X (gfx1250) — compile-verified
//
#include <hip/hip_runtime.h>
#include <hip/hip_bf16.h>
#include <cstdint>

// ---- static problem config (matches reference setup_inputs) ----
#define HP 24
#define WP 24
#define TF 9
#define BB 4
#define NN 50
#define DD 768
#define LL (HP * WP)   // 576 = MAX_ROI_LEN
#define HW (HP * WP)   // 576 spatial positions
#define D4 (DD / 4)    // 192 float4 per row
#define ROWS_PER_BLOCK 8

#if __has_builtin(__builtin_amdgcn_global_load_async_to_lds_b128)
#define HAVE_ASYNC 1
#endif

typedef int v4i __attribute__((ext_vector_type(4)));
typedef __attribute__((address_space(1))) v4i* gv4i_ptr;   // global
typedef __attribute__((address_space(3))) v4i* lv4i_ptr;   // LDS

static __device__ __forceinline__ void wait_async0() {
#if __has_builtin(__builtin_amdgcn_s_wait_asynccnt)
    __builtin_amdgcn_s_wait_asynccnt(0);
#else
    asm volatile("s_wait_asynccnt 0" ::: "memory");
#endif
}

static __device__ __forceinline__ int imin(int a, int b) { return a < b ? a : b; }
static __device__ __forceinline__ int imax(int a, int b) { return a > b ? a : b; }

// ---------------- Kernel A: temporal mean over TF frames ----------------
// sp[b, hw, d] = mean_t x[b, hw*TF + t, d];  one float4 per thread.
__global__ __launch_bounds__(256) void temporal_mean_kernel(
    const float* __restrict__ x, float* __restrict__ sp) {
    const int idx = blockIdx.x * 256 + threadIdx.x;   // over BB*HW*D4 (exact)
    const int d4  = idx % D4;
    const int bhw = idx / D4;
    const float4* p = reinterpret_cast<const float4*>(x) + (size_t)bhw * (TF * D4) + d4;
    float sx = 0.f, sy = 0.f, sz = 0.f, sw = 0.f;
#pragma unroll
    for (int t = 0; t < TF; ++t) {
        float4 v = p[(size_t)t * D4];
        sx += v.x; sy += v.y; sz += v.z; sw += v.w;
    }
    float4 r;
    r.x = sx / 9.0f; r.y = sy / 9.0f; r.z = sz / 9.0f; r.w = sw / 9.0f;
    reinterpret_cast<float4*>(sp)[idx] = r;
}

// ---------------- Kernel B: per-box ROI bounds ----------------
// Replicates the reference's truncation-toward-zero int math exactly.
__global__ __launch_bounds__(256) void box_kernel(
    const float* __restrict__ bb, int4* __restrict__ params) {
    const int i = blockIdx.x * 256 + threadIdx.x;
    if (i >= BB * NN) return;
    const float cx = bb[i * 4 + 0], cy = bb[i * 4 + 1];
    const float bw = bb[i * 4 + 2], bh = bb[i * 4 + 3];
    const float bwp = bw * 1.2f;   // 1 + 2*PADDING
    const float bhp = bh * 1.2f;

    int cs = imax((int)((cx - bwp * 0.5f) * (float)WP), 0);
    int ce = imin((int)((cx + bwp * 0.5f) * (float)WP), WP);
    int rs = imax((int)((cy - bhp * 0.5f) * (float)HP), 0);
    int re = imin((int)((cy + bhp * 0.5f) * (float)HP), HP);

    const int cm = (int)((float)(cs + ce) * 0.5f);  // trunc toward zero, like astype(int32)
    const int rm = (int)((float)(rs + re) * 0.5f);
    // half_min = MIN_PATCHES//2 = 1
    cs = imin(cs, imax(cm - 1, 0));
    ce = imax(ce, imin(cm + 2, WP));
    rs = imin(rs, imax(rm - 1, 0));
    re = imax(re, imin(rm + 2, HP));

    const int h = imax(re - rs, 0);
    const int w = imax(ce - cs, 0);
    params[i] = make_int4(rs, cs, h, w);
}

// ---------------- Kernel C: ROI gather (one output row per wave) ----------------
// grid = BB*NN * (LL/ROWS_PER_BLOCK) blocks of 256 threads (8 wave32 each).
// Wave w of a block handles k = kbase + w for one (b,n): copies one 3KB sp row
// (staged through LDS via the async engine when available) or writes zeros.
__global__ __launch_bounds__(256) void roi_gather_kernel(
    const float* __restrict__ sp, const int4* __restrict__ params,
    float* __restrict__ outF, float* __restrict__ outM) {
#ifdef HAVE_ASYNC
    __shared__ float ldsbuf[ROWS_PER_BLOCK * DD];   // 24 KB
#endif
    const int wv    = threadIdx.x >> 5;
    const int lane  = threadIdx.x & 31;
    const int bid   = blockIdx.x;
    const int bn    = bid / (LL / ROWS_PER_BLOCK);
    const int kbase = (bid % (LL / ROWS_PER_BLOCK)) * ROWS_PER_BLOCK;
    const int k     = kbase + wv;

    const int4 pr = params[bn];
    const int rs = pr.x, cs = pr.y, h = pr.z, w = pr.w;
    const int wsafe = (w > 0) ? w : 1;
    const bool valid = k < h * w;

    float4* dst = reinterpret_cast<float4*>(outF) + (size_t)(bn * LL + k) * D4;

    if (valid) {
        const int srow = rs + k / wsafe;
        const int scol = cs + k % wsafe;
        const int b    = bn / NN;
        const float4* src = reinterpret_cast<const float4*>(sp)
                          + (size_t)(b * HW + srow * WP + scol) * D4;
#ifdef HAVE_ASYNC
        float* myrow = &ldsbuf[wv * DD];
#pragma unroll
        for (int i = 0; i < 6; ++i) {
            const int c = lane + 32 * i;
            __builtin_amdgcn_global_load_async_to_lds_b128(
                (gv4i_ptr)(src + c),
                (lv4i_ptr)(myrow + 4 * c),
                0, 0);
        }
        wait_async0();
        const float4* lrow = reinterpret_cast<const float4*>(myrow);
#pragma unroll
        for (int i = 0; i < 6; ++i) {
            const int c = lane + 32 * i;
            dst[c] = lrow[c];
        }
#else
#pragma unroll
        for (int i = 0; i < 6; ++i) {
            const int c = lane + 32 * i;
            dst[c] = src[c];
        }
#endif
    } else {
        const float4 z = make_float4(0.f, 0.f, 0.f, 0.f);
#pragma unroll
        for (int i = 0; i < 6; ++i) dst[lane + 32 * i] = z;
    }
    if (lane == 0) outM[bn * LL + k] = valid ? 0.0f : 1.0f;  // True (1.0) = padding
}

extern "C" void kernel_launch(void* const* d_in, const int* in_sizes, int n_in,
                              void* d_out, int out_size, void* d_ws, size_t ws_size,
                              hipStream_t stream) {
    const float* x  = (const float*)d_in[0];   // [BB, HW*TF, DD] f32
    const float* bb = (const float*)d_in[1];   // [BB, NN, 4]    f32

    float* outF = (float*)d_out;
    // mask sits at the tail of the concatenated outputs
    float* outM = (float*)d_out + ((size_t)out_size - (size_t)BB * NN * LL);

    // workspace: sp (7,077,888 B) then 200 int4 box params
    float* sp    = (float*)d_ws;
    int4* params = (int4*)((char*)d_ws + (size_t)BB * HW * DD * sizeof(float));

    temporal_mean_kernel<<<(BB * HW * D4) / 256, 256, 0, stream>>>(x, sp);
    box_kernel<<<1, 256, 0, stream>>>(bb, params);
    roi_gather_kernel<<<(BB * NN) * (LL / ROWS_PER_BLOCK), 256, 0, stream>>>(
        sp, params, outF, outM);
}